// WKPooling_61976378081900
// MI455X (gfx1250) — compile-verified
//
#include <hip/hip_runtime.h>
#include <hip/hip_bf16.h>

typedef __attribute__((ext_vector_type(2))) float v2f;
typedef __attribute__((ext_vector_type(8))) float v8f;

#define NLAYERS 12
#define LAYER_START 4
#define TMAX 512
#define HID 768
#define NB 8
#define WKEPS 1e-8f

// ---------------- Kernel 0: T = sum(mask[0,:]) - 1 ----------------
__global__ __launch_bounds__(256) void k0_T(const int* __restrict__ mask,
                                            int* __restrict__ wsT) {
    __shared__ int red[256];
    int tid = threadIdx.x;
    red[tid] = mask[tid] + mask[tid + 256];
    __syncthreads();
    for (int off = 128; off > 0; off >>= 1) {
        if (tid < off) red[tid] += red[tid + off];
        __syncthreads();
    }
    if (tid == 0) *wsT = red[0] - 1;
}

// ---------------- Kernel 1: per-token 8x8 Gram via WMMA f32 16x16x4 ----------
// One wave handles TWO tokens packed as a 16x768 A-matrix:
//   rows 0..7  = token0 layers 0..7, rows 8..15 = token1 layers 0..7.
// Same registers serve as A (16x4) and B (4x16) since the f32 layouts are
// transposes of each other; D's diagonal 8x8 blocks are the two Grams.
__global__ __launch_bounds__(256) void k1_gram(const float* __restrict__ hs,
                                               const int* __restrict__ wsT,
                                               float* __restrict__ Gws) {
    const int T = *wsT;
    const int lane  = threadIdx.x & 31;
    const int wid   = threadIdx.x >> 5;
    const int pair  = blockIdx.x * 8 + wid;      // 2048 pairs total
    const int b     = pair >> 8;                 // 256 pairs per batch
    const int p     = pair & 255;
    const int t0    = p * 2;
    if (t0 >= T) return;                          // wave-uniform exit

    const int row   = lane & 15;                  // M index 0..15
    const int khalf = lane >> 4;                  // 0: K=0,1  1: K=2,3
    const int tsel  = row >> 3;                   // which token
    const int layer = row & 7;
    int t = t0 + tsel;
    if (t >= T) t = t0;                           // clamp (garbage never stored)

    const float* ptr = hs
        + ((size_t)((b * NLAYERS + LAYER_START + layer) * TMAX + t)) * HID
        + khalf * 2;

    v8f acc0 = {0.f,0.f,0.f,0.f,0.f,0.f,0.f,0.f};
    v8f acc1 = {0.f,0.f,0.f,0.f,0.f,0.f,0.f,0.f};
    #pragma unroll 4
    for (int c = 0; c < HID / 4; c += 2) {
        v2f a0, a1;
        a0.x = ptr[0]; a0.y = ptr[1];             // chunk c   (K=4c+2*khalf,+1)
        a1.x = ptr[4]; a1.y = ptr[5];             // chunk c+1
        acc0 = __builtin_amdgcn_wmma_f32_16x16x4_f32(
            false, a0, false, a0, (short)0, acc0, false, false);
        acc1 = __builtin_amdgcn_wmma_f32_16x16x4_f32(
            false, a1, false, a1, (short)0, acc1, false, false);
        ptr += 8;
    }
    v8f acc = acc0 + acc1;

    // D layout: lane<16, VGPR j -> (M=j, N=lane); lane>=16, VGPR j -> (M=8+j, N=lane-16)
    int tok = -1, col = 0;
    if (lane < 8)                      { tok = t0;     col = lane;      }  // G0[j][col]
    else if (lane >= 24 && t0 + 1 < T) { tok = t0 + 1; col = lane - 24; }  // G1[j][col]
    if (tok >= 0) {
        float* Gp = Gws + (size_t)(b * TMAX + tok) * 64;
        #pragma unroll
        for (int j = 0; j < 8; ++j) Gp[j * 8 + col] = acc[j];
    }
}

// ---------------- Kernel 2: per-token alphas (Cholesky == QR's R) + variance --
__global__ __launch_bounds__(256) void k2_alpha(const float* __restrict__ Gws,
                                                const int* __restrict__ wsT,
                                                float* __restrict__ vws,
                                                float* __restrict__ aws) {
    const int id = blockIdx.x * blockDim.x + threadIdx.x;
    const int b = id >> 9;
    const int t = id & 511;
    const int T = *wsT;
    if (t >= T) return;
    const float* G = Gws + (size_t)(b * TMAX + t) * 64;

    // ---- token variance (unnormalized; K3 divides by the per-batch sum) ----
    float nrm[8];
    #pragma unroll
    for (int i = 0; i < 8; ++i) nrm[i] = sqrtf(G[i * 8 + i]);
    float s[7], mean = 0.f;
    #pragma unroll
    for (int i = 0; i < 7; ++i) {
        s[i] = G[(i + 1) * 8 + i] / fmaxf(nrm[i] * nrm[i + 1], WKEPS);
        mean += s[i];
    }
    mean *= (1.0f / 7.0f);
    float var = 0.f;
    #pragma unroll
    for (int i = 0; i < 7; ++i) { float d = s[i] - mean; var += d * d; }
    var *= (1.0f / 6.0f);                           // ddof=1
    vws[b * TMAX + t] = var;

    // ---- per-k align / novelty from Cholesky of Gram submatrix ----
    const int wlen[8]   = {3, 3, 5, 5, 5, 5, 4, 3};
    const int idxt[8][5] = {
        {1, 2, 0, 0, 0}, {2, 3, 1, 0, 0},
        {0, 1, 3, 4, 2}, {1, 2, 4, 5, 3},
        {2, 3, 5, 6, 4}, {3, 4, 6, 7, 5},
        {4, 5, 7, 6, 0}, {5, 6, 7, 0, 0}};

    float al[8], nv[8];
    for (int k = 0; k < 8; ++k) {
        const int w = wlen[k];
        const int* ix = idxt[k];
        float Lm[5][5];
        for (int i = 0; i < w; ++i) {
            for (int j = 0; j <= i; ++j) {
                float sum = G[ix[i] * 8 + ix[j]];
                for (int m = 0; m < j; ++m) sum -= Lm[i][m] * Lm[j][m];
                if (j == i) Lm[i][i] = sqrtf(fmaxf(sum, 1e-30f));
                else        Lm[i][j] = sum / Lm[j][j];
            }
        }
        // R = L^T ; sub = R[:-1,:-1]; coln[j] = sqrt(G[ix[j]][ix[j]])
        float dot = 0.f;
        for (int i = 0; i < w - 1; ++i) {
            float mi = 0.f;
            for (int j = i; j < w - 1; ++j)
                mi += Lm[j][i] / sqrtf(G[ix[j] * 8 + ix[j]]);
            mi *= 1.0f / (float)(w - 1);
            dot += mi * Lm[w - 1][i];               // m @ R[:-1,-1]
        }
        float nh = 0.f;
        for (int i = 0; i < w - 1; ++i) nh += Lm[w - 1][i] * Lm[w - 1][i];
        float align = dot / sqrtf(nh);
        align = 1.0f / (align * (float)w * 2.0f);
        float nov = Lm[w - 1][w - 1] / sqrtf(G[ix[w - 1] * 8 + ix[w - 1]]);
        al[k] = align;
        nv[k] = nov;
    }
    float sa = 0.f, sn = 0.f;
    #pragma unroll
    for (int k = 0; k < 8; ++k) { sa += al[k]; sn += nv[k]; }
    float alpha[8], ssum = 0.f;
    #pragma unroll
    for (int k = 0; k < 8; ++k) { alpha[k] = al[k] / sa + nv[k] / sn; ssum += alpha[k]; }
    float* ap = aws + (size_t)(b * TMAX + t) * 8;
    #pragma unroll
    for (int k = 0; k < 8; ++k) ap[k] = alpha[k] / ssum;
}

// ---------------- Kernel 3: out[b] = sum_{t,l} (v_t * alpha_{t,l} / V_b) * sf ----
__global__ __launch_bounds__(256) void k3_pool(const float* __restrict__ hs,
                                               const int* __restrict__ wsT,
                                               const float* __restrict__ vws,
                                               const float* __restrict__ aws,
                                               float* __restrict__ out) {
    const int b = blockIdx.x;
    const int tid = threadIdx.x;
    const int T = *wsT;
    __shared__ float red[256];
    __shared__ float cf[256];

    float sum = 0.f;
    for (int t = tid; t < TMAX; t += 256)
        if (t < T) sum += vws[b * TMAX + t];
    red[tid] = sum;
    __syncthreads();
    for (int off = 128; off > 0; off >>= 1) {
        if (tid < off) red[tid] += red[tid + off];
        __syncthreads();
    }
    const float invV = 1.0f / red[0];

    float acc0 = 0.f, acc1 = 0.f, acc2 = 0.f;
    for (int tb = 0; tb < TMAX; tb += 32) {
        __syncthreads();
        {   // stage 32 tokens x 8 layers of coefficients into LDS
            int t = tb + (tid >> 3);
            int l = tid & 7;
            float c = 0.f;
            if (t < T)
                c = vws[b * TMAX + t] * invV * aws[(size_t)(b * TMAX + t) * 8 + l];
            cf[tid] = c;
        }
        __syncthreads();
        for (int tt = 0; tt < 32; ++tt) {
            const float* base = hs
                + ((size_t)((b * NLAYERS + LAYER_START) * TMAX + tb + tt)) * HID;
            #pragma unroll
            for (int l = 0; l < 8; ++l) {
                float c = cf[tt * 8 + l];
                const float* src = base + (size_t)l * TMAX * HID;
                acc0 += c * src[tid];
                acc1 += c * src[tid + 256];
                acc2 += c * src[tid + 512];
            }
        }
    }
    out[b * HID + tid]       = acc0;
    out[b * HID + tid + 256] = acc1;
    out[b * HID + tid + 512] = acc2;
}

extern "C" void kernel_launch(void* const* d_in, const int* in_sizes, int n_in,
                              void* d_out, int out_size, void* d_ws, size_t ws_size,
                              hipStream_t stream) {
    const float* hs   = (const float*)d_in[0];   // (8,12,512,768) f32
    const int*   mask = (const int*)d_in[1];     // (8,512) i32
    float*       out  = (float*)d_out;           // (8,768) f32

    int*   wsT = (int*)d_ws;
    float* Gws = (float*)((char*)d_ws + 64);             // 8*512*64 f32 = 1 MB
    float* vws = Gws + (size_t)NB * TMAX * 64;           // 8*512 f32
    float* aws = vws + (size_t)NB * TMAX;                // 8*512*8 f32

    k0_T    <<<1, 256, 0, stream>>>(mask, wsT);
    k1_gram <<<256, 256, 0, stream>>>(hs, wsT, Gws);     // 2048 waves, 1 token-pair each
    k2_alpha<<<(NB * TMAX) / 256, 256, 0, stream>>>(Gws, wsT, vws, aws);
    k3_pool <<<NB, 256, 0, stream>>>(hs, wsT, vws, aws, out);
}